// HOGAT_81767587381367
// MI455X (gfx1250) — compile-verified
//
#include <hip/hip_runtime.h>
#include <math.h>

#define NNODES 20000
#define NEDGES 320000
#define HEADS  8
#define CH     32
#define DIM    256          // HEADS*CH
#define ETOT   (NEDGES + NNODES)   // edges + self loops

typedef __attribute__((ext_vector_type(2))) float v2f;
typedef __attribute__((ext_vector_type(8))) float v8f;

// ---------------------------------------------------------------------------
// WMMA fp32 GEMM:  C[N,256] = A[N, NSEG*256] @ W[NSEG*256, 256] (+bias)(+elu)
// A is given as up to 3 contiguous segments of width 256 (for the concat GEMM).
// One wave computes one 16x16 tile of C, K swept 4 at a time with
// V_WMMA_F32_16X16X4_F32.  NSEG is a template constant so the K loop fully
// unrolls and all load offsets become instruction immediates (IOFFSET):
//   A offsets  <= 1020 B,  B offsets <= 786 KB  (both < 2^23).
//   A-layout (ISA 7.12.2, 32-bit A 16x4): lane M = lane&15,
//     VGPR0/VGPR1 hold K = (lane>>4)*2 + {0,1}  -> one float2 load per step.
//   B-layout (symmetric): lane N = lane&15, VGPR0/1 hold K=(lane>>4)*2+{0,1}.
//   C-layout: VGPR v -> row = tileRow*16 + v + (lane>=16 ? 8:0), col = lane&15.
// ---------------------------------------------------------------------------
template <int NSEG, int ACT>
__global__ __launch_bounds__(256) void hogat_gemm_wmma(
    const float* __restrict__ a0, const float* __restrict__ a1,
    const float* __restrict__ a2, const float* __restrict__ W,
    const float* __restrict__ bias, float* __restrict__ Cmat)
{
  const int lane    = threadIdx.x & 31;
  const int wave    = blockIdx.x * 8 + (threadIdx.x >> 5);
  const int tileRow = wave >> 4;          // DIM/16 = 16 column tiles
  const int tileCol = wave & 15;
  if (tileRow * 16 >= NNODES) return;

  const int l15   = lane & 15;
  const int khalf = (lane >> 4) << 1;     // 0 or 2
  const int arow  = tileRow * 16 + l15;   // A row handled by this lane
  const int col   = tileCol * 16 + l15;   // B/C column handled by this lane

  const float* segs[3] = { a0, a1, a2 };
  v8f acc = {};

  #pragma unroll
  for (int s = 0; s < NSEG; ++s) {
    const float* Ap = segs[s] + (size_t)arow * DIM + khalf;
    const float* Bp = W + ((size_t)(s << 8) + khalf) * DIM + col;
    __builtin_prefetch(Bp, 0, 1);                 // global_prefetch_b8 (L2 warm)
    #pragma unroll
    for (int k0 = 0; k0 < 256; k0 += 4) {
      v2f av = *(const v2f*)(Ap + k0);            // K = khalf, khalf+1
      v2f bv;
      bv.x = Bp[(size_t)k0 * DIM];                // row k0+khalf
      bv.y = Bp[(size_t)k0 * DIM + DIM];          // row k0+khalf+1
      acc = __builtin_amdgcn_wmma_f32_16x16x4_f32(
          /*neg_a=*/false, av, /*neg_b=*/false, bv,
          /*c_mod=*/(short)0, acc, /*reuse_a=*/false, /*reuse_b=*/false);
    }
  }

  const float bval  = bias ? bias[col] : 0.0f;
  const int   rbase = tileRow * 16 + ((lane >> 4) << 3);
  #pragma unroll
  for (int v = 0; v < 8; ++v) {
    float val = acc[v] + bval;
    if (ACT == 1) val = (val > 0.0f) ? val : (expf(val) - 1.0f);   // ELU
    Cmat[(size_t)(rbase + v) * DIM + col] = val;
  }
}

// ---------------------------------------------------------------------------
// Per-(node,head) attention logits: al = sum_c h[n,h,c] * a[h,c]
// ---------------------------------------------------------------------------
__global__ void hogat_attn_coef(const float* __restrict__ h,
                                const float* __restrict__ a_src,
                                const float* __restrict__ a_dst,
                                float* __restrict__ al_src,
                                float* __restrict__ al_dst)
{
  int i = blockIdx.x * blockDim.x + threadIdx.x;
  if (i >= NNODES * HEADS) return;
  int head = i & (HEADS - 1);
  int n    = i >> 3;
  const float* hp = h + (size_t)n * DIM + head * CH;
  const float* as = a_src + head * CH;
  const float* ad = a_dst + head * CH;
  float s = 0.f, d = 0.f;
  #pragma unroll
  for (int c = 0; c < CH; ++c) { float v = hp[c]; s += v * as[c]; d += v * ad[c]; }
  al_src[i] = s;
  al_dst[i] = d;
}

// ---------------------------------------------------------------------------
// Zero the aggregation buffer and reset segment-max / denom
// ---------------------------------------------------------------------------
__global__ void hogat_init(float* __restrict__ agg, unsigned* __restrict__ m_enc,
                           float* __restrict__ denom)
{
  int i = blockIdx.x * blockDim.x + threadIdx.x;
  if (i < NNODES * DIM) agg[i] = 0.0f;
  if (i < NNODES * HEADS) { m_enc[i] = 0u; denom[i] = 0.0f; }
}

// order-preserving float <-> uint map (for atomicMax on floats of any sign)
__device__ __forceinline__ unsigned enc_f(float f) {
  unsigned u = __float_as_uint(f);
  return (u & 0x80000000u) ? ~u : (u | 0x80000000u);
}
__device__ __forceinline__ float dec_f(unsigned u) {
  return (u & 0x80000000u) ? __uint_as_float(u & 0x7fffffffu)
                           : __uint_as_float(~u);
}

__device__ __forceinline__ void edge_sd(const int* ei, int e, int& src, int& dst) {
  if (e < NEDGES) { src = ei[e]; dst = ei[NEDGES + e]; }
  else            { src = dst = e - NEDGES; }            // self loop
}

// ---------------------------------------------------------------------------
// Pass 1: segment max of leaky_relu(al_src[src]+al_dst[dst]) over dst
// ---------------------------------------------------------------------------
__global__ void hogat_edge_max(const int* __restrict__ ei,
                               const float* __restrict__ al_src,
                               const float* __restrict__ al_dst,
                               unsigned* __restrict__ m_enc)
{
  int i = blockIdx.x * blockDim.x + threadIdx.x;
  if (i >= ETOT * HEADS) return;
  int e = i >> 3, head = i & (HEADS - 1);
  int src, dst; edge_sd(ei, e, src, dst);
  float ev = al_src[src * HEADS + head] + al_dst[dst * HEADS + head];
  ev = (ev > 0.0f) ? ev : 0.2f * ev;                     // leaky_relu(0.2)
  atomicMax(&m_enc[dst * HEADS + head], enc_f(ev));
}

// ---------------------------------------------------------------------------
// Pass 2: numerators + segment sum of exp(e - max) over dst
// ---------------------------------------------------------------------------
__global__ void hogat_edge_expsum(const int* __restrict__ ei,
                                  const float* __restrict__ al_src,
                                  const float* __restrict__ al_dst,
                                  const unsigned* __restrict__ m_enc,
                                  float* __restrict__ denom,
                                  float* __restrict__ anum)
{
  int i = blockIdx.x * blockDim.x + threadIdx.x;
  if (i >= ETOT * HEADS) return;
  int e = i >> 3, head = i & (HEADS - 1);
  int src, dst; edge_sd(ei, e, src, dst);
  float ev = al_src[src * HEADS + head] + al_dst[dst * HEADS + head];
  ev = (ev > 0.0f) ? ev : 0.2f * ev;
  float ex = expf(ev - dec_f(m_enc[dst * HEADS + head]));
  anum[i] = ex;
  atomicAdd(&denom[dst * HEADS + head], ex);
}

// ---------------------------------------------------------------------------
// Pass 3: agg[dst,:] += alpha * h[src,:]   (one edge per 256-thread block)
// ---------------------------------------------------------------------------
__global__ __launch_bounds__(256) void hogat_edge_agg(
    const int* __restrict__ ei, const float* __restrict__ h,
    const float* __restrict__ anum, const float* __restrict__ denom,
    float* __restrict__ agg)
{
  int e = blockIdx.x;
  if (e >= ETOT) return;
  int d = threadIdx.x;
  int src, dst; edge_sd(ei, e, src, dst);
  int head = d >> 5;
  float alpha = anum[e * HEADS + head] /
                (denom[dst * HEADS + head] + 1e-16f);
  atomicAdd(&agg[(size_t)dst * DIM + d], h[(size_t)src * DIM + d] * alpha);
}

// ---------------------------------------------------------------------------
// x_hop = agg + bias ; gate = softmax(leaky_relu(x_hop)) ; hop = gate * x_hop
// one node per 256-thread block, LDS tree reductions
// ---------------------------------------------------------------------------
__global__ __launch_bounds__(256) void hogat_bias_gate(
    const float* __restrict__ agg, const float* __restrict__ bias,
    float* __restrict__ hop)
{
  __shared__ float red[256];
  int n = blockIdx.x, t = threadIdx.x;
  float v = agg[(size_t)n * DIM + t] + bias[t];
  float g = (v > 0.0f) ? v : 0.2f * v;
  red[t] = g; __syncthreads();
  for (int s = 128; s > 0; s >>= 1) {
    if (t < s) red[t] = fmaxf(red[t], red[t + s]);
    __syncthreads();
  }
  float mx = red[0]; __syncthreads();
  float ex = expf(g - mx);
  red[t] = ex; __syncthreads();
  for (int s = 128; s > 0; s >>= 1) {
    if (t < s) red[t] += red[t + s];
    __syncthreads();
  }
  hop[(size_t)n * DIM + t] = (ex / red[0]) * v;
}

// ---------------------------------------------------------------------------
// out = sigmoid(x @ Wo + bo), Wo:[256,5]
// ---------------------------------------------------------------------------
__global__ void hogat_final(const float* __restrict__ x,
                            const float* __restrict__ Wo,
                            const float* __restrict__ bo,
                            float* __restrict__ out)
{
  int i = blockIdx.x * blockDim.x + threadIdx.x;
  if (i >= NNODES * 5) return;
  int n = i / 5, o = i % 5;
  float s = bo[o];
  const float* xr = x + (size_t)n * DIM;
  #pragma unroll 8
  for (int k = 0; k < DIM; ++k) s += xr[k] * Wo[k * 5 + o];
  out[i] = 1.0f / (1.0f + expf(-s));
}

// ---------------------------------------------------------------------------
extern "C" void kernel_launch(void* const* d_in, const int* in_sizes, int n_in,
                              void* d_out, int out_size, void* d_ws, size_t ws_size,
                              hipStream_t stream)
{
  const float* x   = (const float*)d_in[0];
  const int*   ei  = (const int*)  d_in[1];
  const float* W0  = (const float*)d_in[2];
  const float* as0 = (const float*)d_in[3];
  const float* ad0 = (const float*)d_in[4];
  const float* b0  = (const float*)d_in[5];
  const float* W1  = (const float*)d_in[6];
  const float* as1 = (const float*)d_in[7];
  const float* ad1 = (const float*)d_in[8];
  const float* b1  = (const float*)d_in[9];
  const float* Wg  = (const float*)d_in[10];
  const float* bg  = (const float*)d_in[11];
  const float* Wo  = (const float*)d_in[12];
  const float* bo  = (const float*)d_in[13];
  float* out = (float*)d_out;

  const size_t ND  = (size_t)NNODES * DIM;     // 5.12M floats
  const size_t NH  = (size_t)NNODES * HEADS;   // 160K
  const size_t ETH = (size_t)ETOT * HEADS;     // 2.72M

  float* ws    = (float*)d_ws;
  float* h     = ws;                  // [N,256] GEMM result / scratch
  float* agg   = ws + ND;             // [N,256] edge aggregation / layer-1 out
  float* hop1  = ws + 2 * ND;
  float* hop2  = ws + 3 * ND;
  float* xbuf  = ws + 4 * ND;         // layer-0 output
  float* al_s  = ws + 5 * ND;
  float* al_d  = al_s + NH;
  float* denom = al_d + NH;
  unsigned* m_enc = (unsigned*)(denom + NH);
  float* anum  = (float*)(m_enc + NH);          // [ETOT, HEADS]
  (void)ws_size; (void)in_sizes; (void)n_in; (void)out_size;

  const int gemm_blocks = ((NNODES / 16) * (DIM / 16) + 7) / 8;  // 8 waves/blk
  const int eh_blocks   = (int)((ETH + 255) / 256);
  const int nh_blocks   = (int)((NH + 255) / 256);
  const int nd_blocks   = (int)((ND + 255) / 256);

  const float* Ws[2]  = { W0,  W1  };
  const float* ass[2] = { as0, as1 };
  const float* ads[2] = { ad0, ad1 };
  const float* bs[2]  = { b0,  b1  };

  const float* xin = x;
  for (int li = 0; li < 2; ++li) {
    const float* hop0  = xin;
    const float* hprev = hop0;
    float* hops[2] = { hop1, hop2 };

    for (int k = 0; k < 2; ++k) {
      // h = hprev @ W   (no bias, no activation)
      hipLaunchKernelGGL((hogat_gemm_wmma<1, 0>), dim3(gemm_blocks), dim3(256), 0, stream,
                         hprev, hprev, hprev, Ws[li], (const float*)nullptr, h);
      hipLaunchKernelGGL(hogat_attn_coef, dim3(nh_blocks), dim3(256), 0, stream,
                         h, ass[li], ads[li], al_s, al_d);
      hipLaunchKernelGGL(hogat_init, dim3(nd_blocks), dim3(256), 0, stream,
                         agg, m_enc, denom);
      hipLaunchKernelGGL(hogat_edge_max, dim3(eh_blocks), dim3(256), 0, stream,
                         ei, al_s, al_d, m_enc);
      hipLaunchKernelGGL(hogat_edge_expsum, dim3(eh_blocks), dim3(256), 0, stream,
                         ei, al_s, al_d, m_enc, denom, anum);
      hipLaunchKernelGGL(hogat_edge_agg, dim3(ETOT), dim3(256), 0, stream,
                         ei, h, anum, denom, agg);
      hipLaunchKernelGGL(hogat_bias_gate, dim3(NNODES), dim3(256), 0, stream,
                         agg, bs[li], hops[k]);
      hprev = hops[k];
    }

    // x = concat(hop0,hop1,hop2) @ Wg + bg  (+ELU after layer 0)
    float* xo = (li == 0) ? xbuf : agg;   // never aliases the A segments
    if (li == 0) {
      hipLaunchKernelGGL((hogat_gemm_wmma<3, 1>), dim3(gemm_blocks), dim3(256), 0, stream,
                         hop0, hop1, hop2, Wg, bg, xo);
    } else {
      hipLaunchKernelGGL((hogat_gemm_wmma<3, 0>), dim3(gemm_blocks), dim3(256), 0, stream,
                         hop0, hop1, hop2, Wg, bg, xo);
    }
    xin = xo;
  }

  hipLaunchKernelGGL(hogat_final, dim3((NNODES * 5 + 255) / 256), dim3(256), 0, stream,
                     xin, Wo, bo, out);
}